// PhotonInterferenceSeq2Seq_1984274890956
// MI455X (gfx1250) — compile-verified
//
#include <hip/hip_runtime.h>

typedef __attribute__((ext_vector_type(2))) float v2f;
typedef __attribute__((ext_vector_type(4))) float v4f;
typedef __attribute__((ext_vector_type(8))) float v8f;

#define VOCAB 67
#define EMB   50
#define NLAYER 8
#define MP 80      // 67 rows padded to 5 x 16
#define NP 64      // 50 feature cols padded to 4 x 16
#define TPB1 640   // 20 waves (wave32): one 16x16 tile per wave in the layer phase

__device__ __forceinline__ v8f wmma4(v2f a, v2f b, v8f c) {
  // D(16x16,f32) = A(16x4,f32) x B(4x16,f32) + C
  return __builtin_amdgcn_wmma_f32_16x16x4_f32(
      /*neg_a=*/false, a, /*neg_b=*/false, b,
      /*c_mod=*/(short)0, c, /*reuse_a=*/false, /*reuse_b=*/false);
}

// Branch-free zero-padded load: unconditional load from a clamped (valid)
// address + v_cndmask select. Avoids s_and_saveexec around every global load.
__device__ __forceinline__ float ld_pad(const float* __restrict__ base,
                                        int r, int c, int R, int C) {
  const int rc = r < R ? r : R - 1;
  const int cc = c < C ? c : C - 1;
  const float v = base[rc * C + cc];
  return (r < R && c < C) ? v : 0.f;
}

// One workgroup: run the whole 8-layer complex pipeline + output projection for
// all 67 vocabulary entries, producing logits_table[67][67].
__global__ __launch_bounds__(TPB1) void photon_table_kernel(
    const float* __restrict__ emb,    // [67,50]
    const float* __restrict__ kappa,  // [8,50,50]
    const float* __restrict__ phi,    // [8,50,50]
    const float* __restrict__ Wout,   // [50,67]
    const float* __restrict__ bout,   // [67]
    float* __restrict__ table) {      // [67,67]
  __shared__ float sAr[MP * NP];  // 20 KB
  __shared__ float sAi[MP * NP];  // 20 KB
  const int tid  = threadIdx.x;
  const int wave = tid >> 5;
  const int lane = tid & 31;
  const bool hi  = lane >= 16;    // lanes 16..31 hold K+2/K+3 (A,B) and M+8 (C/D)
  const int l16  = lane & 15;

  // Zero-padded load of embedding into A_real; A_imag = 0.
  for (int i = tid; i < MP * NP; i += TPB1) {
    const int r = i >> 6, c = i & 63;
    sAr[i] = ld_pad(emb, r, c, VOCAB, EMB);
    sAi[i] = 0.f;
  }
  __syncthreads();

  // Tile assignment for the layer phase: 5 row-tiles x 4 col-tiles = 20 waves.
  const int m0   = (wave >> 2) << 4;
  const int n0   = (wave & 3) << 4;
  const int mrow = m0 + l16;   // A-fragment row (this lane)
  const int ncol = n0 + l16;   // B-fragment col (this lane)

  for (int L = 0; L < NLAYER; ++L) {
    const float* Kw = kappa + L * EMB * EMB;
    const float* Pw = phi   + L * EMB * EMB;
    v8f nr = {}; v8f ni = {};
#pragma unroll
    for (int k0 = 0; k0 < 52; k0 += 4) {           // K = 50 padded to 52
      const int ka = k0 + (hi ? 2 : 0);
      v2f ar, ai;
      ar.x = sAr[mrow * NP + ka]; ar.y = sAr[mrow * NP + ka + 1];  // cols >=50 are 0
      ai.x = sAi[mrow * NP + ka]; ai.y = sAi[mrow * NP + ka + 1];
      v2f bk, bp;
      bk.x = ld_pad(Kw, ka,     ncol, EMB, EMB);
      bk.y = ld_pad(Kw, ka + 1, ncol, EMB, EMB);
      bp.x = ld_pad(Pw, ka,     ncol, EMB, EMB);
      bp.y = ld_pad(Pw, ka + 1, ncol, EMB, EMB);
      v2f nai; nai.x = -ai.x; nai.y = -ai.y;       // f32 WMMA has no A-negate modifier
      nr = wmma4(ar,  bk, nr);                     // + ar @ kappa
      nr = wmma4(nai, bp, nr);                     // - ai @ phi
      ni = wmma4(ar,  bp, ni);                     // + ar @ phi
      ni = wmma4(ai,  bk, ni);                     // + ai @ kappa
    }
    __syncthreads();   // all waves finished reading sAr/sAi for this layer
#pragma unroll
    for (int j = 0; j < 8; ++j) {                  // C/D layout: VGPR j -> M=j (lo) / j+8 (hi)
      const int row = m0 + j + (hi ? 8 : 0);
      sAr[row * NP + ncol] = nr[j];
      sAi[row * NP + ncol] = ni[j];
    }
    __syncthreads();   // new layer state visible
  }

  // intensity = ar^2 + ai^2, in place in sAr
  for (int i = tid; i < MP * NP; i += TPB1) {
    const float r = sAr[i], im = sAi[i];
    sAr[i] = r * r + im * im;
  }
  __syncthreads();

  // logits_table = intensity @ W_out + b_out : 5 row-tiles x 5 col-tiles (67 -> 80)
  for (int t = wave; t < 25; t += 20) {
    const int pm0 = (t / 5) * 16;
    const int pn0 = (t % 5) * 16;
    const int prow = pm0 + l16;
    const int pcol = pn0 + l16;
    v8f acc = {};
#pragma unroll
    for (int k0 = 0; k0 < 52; k0 += 4) {
      const int ka = k0 + (hi ? 2 : 0);
      v2f a;
      a.x = sAr[prow * NP + ka]; a.y = sAr[prow * NP + ka + 1];
      v2f b;
      b.x = ld_pad(Wout, ka,     pcol, EMB, VOCAB);
      b.y = ld_pad(Wout, ka + 1, pcol, EMB, VOCAB);
      acc = wmma4(a, b, acc);
    }
    if (pcol < VOCAB) {
      const float bb = bout[pcol];
#pragma unroll
      for (int j = 0; j < 8; ++j) {
        const int row = pm0 + j + (hi ? 8 : 0);
        if (row < VOCAB) table[row * VOCAB + pcol] = acc[j] + bb;
      }
    }
  }
}

// out[b,s,v] = table[seq[b,s], v] — pure store-bandwidth kernel (~140 MB writes).
// Each thread emits one aligned 16B nontemporal store (global_store_b128).
__global__ __launch_bounds__(256) void photon_gather_kernel(
    const int* __restrict__ seq,      // [B*S]
    const float* __restrict__ table,  // [67,67], lives in L1/L2
    float* __restrict__ out,          // [B*S*67]
    unsigned total) {
  const unsigned gid = blockIdx.x * blockDim.x + threadIdx.x;
  const unsigned e   = gid * 4u;
  if (e + 3u < total) {
    v4f v;
#pragma unroll
    for (int i = 0; i < 4; ++i) {
      const unsigned ee = e + (unsigned)i;
      const unsigned r  = ee / (unsigned)VOCAB;   // mul+shift
      const unsigned c  = ee - r * (unsigned)VOCAB;
      v[i] = table[(unsigned)seq[r] * (unsigned)VOCAB + c];
    }
    __builtin_nontemporal_store(v, (v4f*)(out + e));
  } else if (e < total) {
    for (unsigned ee = e; ee < total; ++ee) {
      const unsigned r = ee / (unsigned)VOCAB;
      const unsigned c = ee - r * (unsigned)VOCAB;
      __builtin_nontemporal_store(table[(unsigned)seq[r] * (unsigned)VOCAB + c],
                                  out + ee);
    }
  }
}

extern "C" void kernel_launch(void* const* d_in, const int* in_sizes, int n_in,
                              void* d_out, int out_size, void* d_ws, size_t ws_size,
                              hipStream_t stream) {
  const int*   seq   = (const int*)d_in[0];
  const float* emb   = (const float*)d_in[1];
  const float* kappa = (const float*)d_in[2];
  const float* phi   = (const float*)d_in[3];
  const float* Wout  = (const float*)d_in[4];
  const float* bout  = (const float*)d_in[5];
  float* out   = (float*)d_out;
  float* table = (float*)d_ws;   // 67*67*4 = ~18 KB scratch

  photon_table_kernel<<<1, TPB1, 0, stream>>>(emb, kappa, phi, Wout, bout, table);

  const unsigned total   = (unsigned)out_size;          // 64*8192*67
  const unsigned threads = (total + 3u) / 4u;           // one float4 per thread
  photon_gather_kernel<<<(threads + 255u) / 256u, 256, 0, stream>>>(seq, table, out, total);
}